// FocalTverskyLoss_2241972928954
// MI455X (gfx1250) — compile-verified
//
#include <hip/hip_runtime.h>
#include <hip/hip_bf16.h>

// ---------------------------------------------------------------------------
// Focal Tversky loss, CDNA5 (gfx1250, wave32).
//   input  : float32 [16, 19, 512, 512]  (logits)
//   target : int32   [16, 512, 512]      (class ids, harness ABI: int*)
//   out    : float32 [1]
// Memory-bound streamer (~335 MB @ 23.3 TB/s ~ 14.4 us floor).
// WMMA f32 16x16x4 used for the block-level 256x19 column-sum reduction.
// ---------------------------------------------------------------------------

typedef __attribute__((ext_vector_type(4)))  float fl4;
typedef __attribute__((ext_vector_type(4)))  int   in4;
typedef __attribute__((ext_vector_type(2)))  float v2f;
typedef __attribute__((ext_vector_type(8)))  float v8f;

#define CLS      19
#define CPAD     32
#define NB       16
#define HWTOT    (512 * 512)
#define BC       (NB * CLS)
#define TPB      256
#define PXT      4
#define ITERS    8
#define PIXBLK   (TPB * PXT * ITERS)      /* 8192 pixels per block       */
#define CHUNKS   (HWTOT / PIXBLK)         /* 32 chunks per batch image   */
#define ROWP     33                       /* LDS row pitch (bank-safe)   */

__device__ __forceinline__ fl4 max4(fl4 a, fl4 b) {
    fl4 r; r.x = fmaxf(a.x, b.x); r.y = fmaxf(a.y, b.y);
           r.z = fmaxf(a.z, b.z); r.w = fmaxf(a.w, b.w); return r;
}

__global__ void ftl_zero(float* __restrict__ g) {
    int i = threadIdx.x;
    if (i < 3 * BC) g[i] = 0.0f;
}

__global__ __launch_bounds__(TPB)
void ftl_main(const float* __restrict__ in, const int* __restrict__ tgt,
              float* __restrict__ g) {
    const int chunk = blockIdx.x;
    const int b     = blockIdx.y;
    const int tid   = threadIdx.x;

    __shared__ float lds_tp[CLS];
    __shared__ float lds_cnt[CLS];
    __shared__ float red[TPB * ROWP];

    if (tid < CLS) { lds_tp[tid] = 0.0f; lds_cnt[tid] = 0.0f; }
    __syncthreads();

    const float* basep = in  + (size_t)b * CLS * HWTOT;
    const int*   tb    = tgt + (size_t)b * HWTOT;

    float s_acc[CLS];
#pragma unroll
    for (int c = 0; c < CLS; ++c) s_acc[c] = 0.0f;

    for (int it = 0; it < ITERS; ++it) {
        const int px = chunk * PIXBLK + it * (TPB * PXT) + tid * PXT;

        fl4 e[CLS];
#pragma unroll
        for (int c = 0; c < CLS; ++c)
            e[c] = __builtin_nontemporal_load(
                       (const fl4*)(basep + (size_t)c * HWTOT + px));
        in4 t = __builtin_nontemporal_load((const in4*)(tb + px));

        // per-pixel max over classes (numerical safety)
        fl4 mx = e[0];
#pragma unroll
        for (int c = 1; c < CLS; ++c) mx = max4(mx, e[c]);

        fl4 S  = {0.0f, 0.0f, 0.0f, 0.0f};
        fl4 et = {0.0f, 0.0f, 0.0f, 0.0f};   // exp of the target-class logit
#pragma unroll
        for (int c = 0; c < CLS; ++c) {
            fl4 ex;
            ex.x = __expf(e[c].x - mx.x);
            ex.y = __expf(e[c].y - mx.y);
            ex.z = __expf(e[c].z - mx.z);
            ex.w = __expf(e[c].w - mx.w);
            S.x += ex.x; S.y += ex.y; S.z += ex.z; S.w += ex.w;
            et.x = (t.x == c) ? ex.x : et.x;
            et.y = (t.y == c) ? ex.y : et.y;
            et.z = (t.z == c) ? ex.z : et.z;
            et.w = (t.w == c) ? ex.w : et.w;
            e[c] = ex;
        }
        fl4 r;
        r.x = __fdividef(1.0f, S.x); r.y = __fdividef(1.0f, S.y);
        r.z = __fdividef(1.0f, S.z); r.w = __fdividef(1.0f, S.w);

#pragma unroll
        for (int c = 0; c < CLS; ++c)
            s_acc[c] += e[c].x * r.x + e[c].y * r.y + e[c].z * r.z + e[c].w * r.w;

        // target-indexed terms via LDS float atomics (ds_add_f32)
        atomicAdd(&lds_tp[t.x], et.x * r.x);
        atomicAdd(&lds_tp[t.y], et.y * r.y);
        atomicAdd(&lds_tp[t.z], et.z * r.z);
        atomicAdd(&lds_tp[t.w], et.w * r.w);
        atomicAdd(&lds_cnt[t.x], 1.0f);
        atomicAdd(&lds_cnt[t.y], 1.0f);
        atomicAdd(&lds_cnt[t.z], 1.0f);
        atomicAdd(&lds_cnt[t.w], 1.0f);
    }

    // deposit per-thread softmax-sum partials as one row of a 256 x 32 matrix
#pragma unroll
    for (int c = 0; c < CPAD; ++c)
        red[tid * ROWP + c] = (c < CLS) ? s_acc[c] : 0.0f;
    __syncthreads();

    if (tid < CLS) {
        atomicAdd(&g[BC     + b * CLS + tid], lds_tp[tid]);
        atomicAdd(&g[2 * BC + b * CLS + tid], lds_cnt[tid]);
    }

    // Wave 0: column-sum the 256 x 32 matrix with f32 WMMA (D = ones * B + C).
    // With A = all-ones the result is invariant to B's row->k slot mapping;
    // only "column n <-> lane n / n+16" matters (matches documented layout).
    if (tid < 32) {
        const int col  = tid & 15;
        const int half = tid >> 4;
        v8f acc0 = {0.f,0.f,0.f,0.f,0.f,0.f,0.f,0.f};
        v8f acc1 = {0.f,0.f,0.f,0.f,0.f,0.f,0.f,0.f};
        v2f ones; ones.x = 1.0f; ones.y = 1.0f;
        for (int r0 = 0; r0 < TPB; r0 += 4) {
            v2f b0, b1;
            b0.x = red[(r0 + half)     * ROWP + col];
            b0.y = red[(r0 + 2 + half) * ROWP + col];
            b1.x = red[(r0 + half)     * ROWP + col + 16];
            b1.y = red[(r0 + 2 + half) * ROWP + col + 16];
            acc0 = __builtin_amdgcn_wmma_f32_16x16x4_f32(
                       false, ones, false, b0, (short)0, acc0, false, false);
            acc1 = __builtin_amdgcn_wmma_f32_16x16x4_f32(
                       false, ones, false, b1, (short)0, acc1, false, false);
        }
        // D[m,n] identical for all m; lane j holds column j&15.
        if (tid < 16)            atomicAdd(&g[b * CLS + tid], acc0[0]);
        else if ((tid - 16) < 3) atomicAdd(&g[b * CLS + 16 + (tid - 16)], acc1[0]);
    }
}

__global__ void ftl_final(const float* __restrict__ g, float* __restrict__ out) {
    const int lane = threadIdx.x;           // one wave32
    float partial = 0.0f;
    for (int i = lane; i < BC; i += 32) {
        float sum_p = g[i];
        float tp    = g[BC + i];
        float cnt   = g[2 * BC + i];
        float fp = sum_p - tp;
        float fn = cnt   - tp;
        float tv = (tp + 1.0f) / (tp + 0.5f * fp + 0.5f * fn + 1.0f);
        partial += (1.0f - tv);             // GAMMA == 1
    }
#pragma unroll
    for (int off = 16; off > 0; off >>= 1)
        partial += __shfl_xor(partial, off, 32);
    if (lane == 0) out[0] = partial / (float)NB;
}

extern "C" void kernel_launch(void* const* d_in, const int* in_sizes, int n_in,
                              void* d_out, int out_size, void* d_ws, size_t ws_size,
                              hipStream_t stream) {
    const float* in  = (const float*)d_in[0];
    const int*   tgt = (const int*)  d_in[1];
    float*       g   = (float*)d_ws;        // [3 * 16 * 19] accumulators
    float*       out = (float*)d_out;

    ftl_zero<<<1, 1024, 0, stream>>>(g);
    dim3 grid(CHUNKS, NB);
    ftl_main<<<grid, TPB, 0, stream>>>(in, tgt, g);
    ftl_final<<<1, 32, 0, stream>>>(g, out);
}